// SparseRetrievalModel_35562329211611
// MI455X (gfx1250) — compile-verified
//
#include <hip/hip_runtime.h>
#include <stdint.h>

// SparseRetrievalModel: scores = CSR(values, indice, 64 nnz/doc) @ x ; top-10.
// Memory-bound: ~103 MB streamed, 25.6 MFLOP -> 0.25 flop/B; HBM floor ~4.4us.
//  - TDM (tensor_load_to_lds) stages the 122 KB query vector x into LDS once
//    per block (fits in 320 KB/WGP LDS) -> random gathers become ds_loads.
//  - One wave32 per doc: lanes load int2/float2 (256 B coalesced per instr),
//    5x shfl_xor butterfly -> every lane holds the doc score.
//  - Scores are NEVER written to memory: each wave keeps an in-register
//    top-10, block merges 8x10 candidates in LDS, emits 10 (val,doc) pairs.
//  - Final kernel merges 512x10 = 5120 candidates (40 KB) -> top-k output.
//  - global_prefetch_b8 for the next doc chunk.

#define NNZ_PER_DOC   64
#define WAVES_PER_BLK 8
#define BLK_THREADS   256
#define VOCAB_C       30522
#define TOPK          10
#define SCORE_BLOCKS  512
#define N_CAND        (SCORE_BLOCKS * TOPK)

typedef unsigned int u32x4 __attribute__((ext_vector_type(4)));
typedef int          i32x8 __attribute__((ext_vector_type(8)));
typedef int          i32x4 __attribute__((ext_vector_type(4)));

#if defined(__has_builtin)
#if __has_builtin(__builtin_amdgcn_tensor_load_to_lds)
#define HAVE_TDM 1
#endif
#endif

__global__ __launch_bounds__(BLK_THREADS) void sr_score_topk_kernel(
    const float* __restrict__ x,
    const int*   __restrict__ idx,
    const float* __restrict__ val,
    float*       __restrict__ cand_v,
    int*         __restrict__ cand_i,
    int n_docs)
{
    // Single LDS object -> base at LDS offset 0 (TDM descriptor lds_addr = 0).
    // Partition: [0, VOCAB_C) = x staging; tail = wave candidate scratch.
    __shared__ float lds[VOCAB_C + 2 * WAVES_PER_BLK * TOPK];
    float* xs = lds;
    float* wv = lds + VOCAB_C;                               // 80 floats
    int*   wi = (int*)(lds + VOCAB_C + WAVES_PER_BLK * TOPK); // 80 ints

#ifdef HAVE_TDM
    // One wave issues the DMA of x[0..30521] into LDS; SGPR-uniform branch so
    // other waves skip the instruction entirely (TDM ignores EXEC).
    if (__builtin_amdgcn_readfirstlane((int)threadIdx.x) < 32) {
        uint64_t ga = (uint64_t)(uintptr_t)x;
        u32x4 g0;
        g0[0] = 1u;                                           // count = 1 valid descriptor
        g0[1] = 0u;                                           // lds_addr = 0 (xs)
        g0[2] = (uint32_t)ga;                                 // global_addr[31:0]
        g0[3] = (uint32_t)((ga >> 32) & 0x1FFFFFFu)           // global_addr[56:32]
              | (2u << 30);                                   // type = 2 ("image")
        i32x8 g1;
        g1[0] = (int)(2u << 16);                              // data_size = 4 bytes
        g1[1] = (int)((VOCAB_C & 0xFFFFu) << 16);             // tensor_dim0[15:0]
        g1[2] = (int)(((uint32_t)VOCAB_C >> 16)               // tensor_dim0[31:16]
              | (1u << 16));                                  // tensor_dim1 = 1 (lo16)
        g1[3] = (int)((VOCAB_C & 0xFFFFu) << 16);             // tensor_dim1 hi=0 | tile_dim0
        g1[4] = 1;                                            // tile_dim1 = 1, tile_dim2 = 0
        g1[5] = VOCAB_C;                                      // tensor_dim0_stride[31:0]
        g1[6] = (int)((VOCAB_C & 0xFFFFu) << 16);             // stride0 hi=0 | stride1[15:0]
        g1[7] = 0;                                            // stride1[47:16] = 0
        i32x4 gz4 = {0, 0, 0, 0};                             // groups 2/3: unused dims
        i32x8 gz8 = {0, 0, 0, 0, 0, 0, 0, 0};                 // extra group (clang-23 form)
        __builtin_amdgcn_tensor_load_to_lds(g0, g1, gz4, gz4, gz8, 0);
        __builtin_amdgcn_s_wait_tensorcnt(0);
    }
#else
    for (int i = threadIdx.x; i < VOCAB_C; i += BLK_THREADS) xs[i] = x[i];
#endif
    __syncthreads();

    const int lane = threadIdx.x & 31;
    const int wave = threadIdx.x >> 5;
    const int gw   = blockIdx.x * WAVES_PER_BLK + wave;
    const int nw   = gridDim.x * WAVES_PER_BLK;

    const int2*   ip = (const int2*)idx;   // 2 nnz per lane -> 256 B coalesced loads
    const float2* vp = (const float2*)val;

    // Per-wave running top-10 (identical in all 32 lanes), sorted descending.
    float bv[TOPK];
    int   bi[TOPK];
    #pragma unroll
    for (int j = 0; j < TOPK; ++j) { bv[j] = -__builtin_huge_valf(); bi[j] = -1; }

    for (int d = gw; d < n_docs; d += nw) {
        const int base = d * (NNZ_PER_DOC / 2) + lane;
        const int nd   = d + nw;
        if (nd < n_docs) {
            __builtin_prefetch(&ip[nd * (NNZ_PER_DOC / 2) + lane], 0, 0);
            __builtin_prefetch(&vp[nd * (NNZ_PER_DOC / 2) + lane], 0, 0);
        }
        const int2   i2 = ip[base];
        const float2 v2 = vp[base];
        float s = v2.x * xs[i2.x] + v2.y * xs[i2.y];
        #pragma unroll
        for (int off = 16; off >= 1; off >>= 1)
            s += __shfl_xor(s, off, 32);          // butterfly: all lanes get sum

        if (s > bv[TOPK - 1]) {                   // wave-uniform condition
            bv[TOPK - 1] = s; bi[TOPK - 1] = d;
            #pragma unroll
            for (int j = TOPK - 1; j > 0; --j) {
                if (bv[j] > bv[j - 1]) {
                    float tv = bv[j]; bv[j] = bv[j - 1]; bv[j - 1] = tv;
                    int   ti = bi[j]; bi[j] = bi[j - 1]; bi[j - 1] = ti;
                }
            }
        }
    }

    if (lane == 0) {
        #pragma unroll
        for (int j = 0; j < TOPK; ++j) {
            wv[wave * TOPK + j] = bv[j];
            wi[wave * TOPK + j] = bi[j];
        }
    }
    __syncthreads();

    // Merge 80 wave candidates -> block top-10 (cheap, once per block).
    if (threadIdx.x == 0) {
        for (int r = 0; r < TOPK; ++r) {
            float best = -__builtin_huge_valf();
            int   bs   = 0;
            for (int c = 0; c < WAVES_PER_BLK * TOPK; ++c)
                if (wv[c] > best) { best = wv[c]; bs = c; }
            cand_v[blockIdx.x * TOPK + r] = best;
            cand_i[blockIdx.x * TOPK + r] = wi[bs];
            wv[bs] = -__builtin_huge_valf();
        }
    }
}

__global__ __launch_bounds__(BLK_THREADS) void sr_topk_final_kernel(
    const float* __restrict__ cand_v_g,
    const int*   __restrict__ cand_i_g,
    int n_cand, int k, float* __restrict__ out)
{
    __shared__ float cv[BLK_THREADS * TOPK];
    __shared__ int   ci[BLK_THREADS * TOPK];
    __shared__ float rv[WAVES_PER_BLK];
    __shared__ int   rs[WAVES_PER_BLK];

    const int t = threadIdx.x;

    float bv[TOPK];
    int   bi[TOPK];
    #pragma unroll
    for (int j = 0; j < TOPK; ++j) { bv[j] = -__builtin_huge_valf(); bi[j] = -1; }

    for (int i = t; i < n_cand; i += BLK_THREADS) {
        const float s = cand_v_g[i];
        if (s > bv[TOPK - 1]) {
            bv[TOPK - 1] = s; bi[TOPK - 1] = cand_i_g[i];
            #pragma unroll
            for (int j = TOPK - 1; j > 0; --j) {
                if (bv[j] > bv[j - 1]) {
                    float tv = bv[j]; bv[j] = bv[j - 1]; bv[j - 1] = tv;
                    int   ti = bi[j]; bi[j] = bi[j - 1]; bi[j - 1] = ti;
                }
            }
        }
    }

    #pragma unroll
    for (int j = 0; j < TOPK; ++j) { cv[t * TOPK + j] = bv[j]; ci[t * TOPK + j] = bi[j]; }
    __syncthreads();

    // k rounds of block-wide argmax over the 2560 LDS candidates.
    for (int r = 0; r < k; ++r) {
        float mv = -__builtin_huge_valf();
        int   ms = -1;
        #pragma unroll
        for (int j = 0; j < TOPK; ++j) {
            const float v = cv[t * TOPK + j];
            if (v > mv) { mv = v; ms = t * TOPK + j; }
        }
        #pragma unroll
        for (int off = 16; off >= 1; off >>= 1) {
            const float ov = __shfl_xor(mv, off, 32);
            const int   os = __shfl_xor(ms, off, 32);
            if (ov > mv) { mv = ov; ms = os; }
        }
        if ((t & 31) == 0) { rv[t >> 5] = mv; rs[t >> 5] = ms; }
        __syncthreads();
        if (t == 0) {
            float fv = -__builtin_huge_valf();
            int   fs = 0;
            #pragma unroll
            for (int w = 0; w < WAVES_PER_BLK; ++w)
                if (rv[w] > fv) { fv = rv[w]; fs = rs[w]; }
            out[r]     = fv;                  // top value r
            out[k + r] = (float)ci[fs];       // top index r (flattened after values)
            cv[fs] = -__builtin_huge_valf();  // remove winner
        }
        __syncthreads();
    }
}

extern "C" void kernel_launch(void* const* d_in, const int* in_sizes, int n_in,
                              void* d_out, int out_size, void* d_ws, size_t ws_size,
                              hipStream_t stream)
{
    const float* x      = (const float*)d_in[0];
    const int*   crow   = (const int*)d_in[1];   // uniform: 64 nnz/doc (unused)
    const int*   indice = (const int*)d_in[2];
    const float* values = (const float*)d_in[3];
    (void)crow; (void)n_in; (void)ws_size;

    const int n_docs = in_sizes[1] - 1;          // crow has n_docs+1 entries
    int k = out_size / 2;                        // tuple (vals[k], idx[k]) flattened
    if (k > TOPK) k = TOPK;

    // Workspace: 5120 candidate values + 5120 candidate doc ids (40 KB).
    float* cand_v = (float*)d_ws;
    int*   cand_i = (int*)d_ws + N_CAND;

    sr_score_topk_kernel<<<SCORE_BLOCKS, BLK_THREADS, 0, stream>>>(
        x, indice, values, cand_v, cand_i, n_docs);
    sr_topk_final_kernel<<<1, BLK_THREADS, 0, stream>>>(
        cand_v, cand_i, N_CAND, k, (float*)d_out);
}